// Line_85323820302554
// MI455X (gfx1250) — compile-verified
//
#include <hip/hip_runtime.h>

// Piecewise-linear LUT kernel for MI455X (gfx1250, wave32).
// Memory-bound streaming op: ~201 MB traffic -> ~8.6 us floor at 23.3 TB/s.
// Strategy: B128 nontemporal global load/store, per-plane LUT staged in LDS
// as (lo,hi) float2 pairs so each element costs one ds_load_b64.

typedef float v4f __attribute__((ext_vector_type(4)));
typedef float v2f __attribute__((ext_vector_type(2)));

#define LUT_K        101          // knots
#define LUT_PAIRS    (LUT_K - 1)  // 100 (lo,hi) pairs
#define HW           (512 * 512)  // elements per (b,c) plane
#define N_BATCH      32
#define N_CHAN       3
#define N_PLANES     (N_BATCH * N_CHAN)      // 96
#define VEC4_PER_PLANE (HW / 4)              // 65536
#define BLOCKS_PER_PLANE 32
#define THREADS      256                     // 8 wave32s

__global__ __launch_bounds__(THREADS)
void lut_lerp_kernel(const float* __restrict__ input,
                     const int*   __restrict__ labels,
                     const float* __restrict__ weightx,
                     const float* __restrict__ weighty,
                     float*       __restrict__ out)
{
    __shared__ v2f lut[LUT_PAIRS];

    const int plane = blockIdx.y;          // 0..95
    const int b = plane / N_CHAN;
    const int c = plane % N_CHAN;
    const int label = labels[b];           // uniform -> s_load

    // Stage LUT as pre-paired (y[i], y[i+1]) so the inner loop does one
    // ds_load_b64 per element instead of two ds_load_b32.
    const float* row = weighty + ((size_t)label * N_CHAN + c) * LUT_K;
    if (threadIdx.x < LUT_PAIRS) {
        v2f p;
        p.x = row[threadIdx.x];
        p.y = row[threadIdx.x + 1];
        lut[threadIdx.x] = p;
    }

    // Uniform grid parameters (uniform addresses -> scalar loads).
    const float x0      = weightx[0];
    const float invStep = 1.0f / (weightx[1] - weightx[0]);

    __syncthreads();

    const v4f* __restrict__ in4  = (const v4f*)(input + (size_t)plane * HW);
    v4f*       __restrict__ out4 = (v4f*)(out + (size_t)plane * HW);

    // Grid-stride over the plane in float4 units: 32 blocks x 256 thr
    // -> 8 iterations/thread, each iteration = 16B in + 16B out per lane.
    for (int i = blockIdx.x * THREADS + threadIdx.x;
         i < VEC4_PER_PLANE;
         i += BLOCKS_PER_PLANE * THREADS)
    {
        // Streamed once: bypass caches (TH=NT) on both sides.
        v4f x = __builtin_nontemporal_load(&in4[i]);
        v4f y;
#pragma unroll
        for (int j = 0; j < 4; ++j) {
            float pos = (x[j] - x0) * invStep;
            float f   = __builtin_floorf(pos);
            // clamp bin to [0, K-2]; NaN-safe (maxf(NaN,0)=0)
            f = fminf(fmaxf(f, 0.0f), (float)(LUT_K - 2));
            int idx = (int)f;
            // clamp fraction -> flat extrapolation at both ends
            float t = fminf(fmaxf(pos - f, 0.0f), 1.0f);
            v2f p = lut[idx];                 // ds_load_b64
            y[j] = __builtin_fmaf(t, p.y - p.x, p.x);
        }
        __builtin_nontemporal_store(y, &out4[i]);
    }
}

extern "C" void kernel_launch(void* const* d_in, const int* in_sizes, int n_in,
                              void* d_out, int out_size, void* d_ws, size_t ws_size,
                              hipStream_t stream)
{
    const float* input   = (const float*)d_in[0];  // [32,3,512,512] f32
    const int*   labels  = (const int*)  d_in[1];  // [32] i32
    const float* weightx = (const float*)d_in[2];  // [101] f32
    const float* weighty = (const float*)d_in[3];  // [25,3,101] f32
    float*       out     = (float*)d_out;          // [32,3,512,512] f32

    dim3 grid(BLOCKS_PER_PLANE, N_PLANES, 1);
    lut_lerp_kernel<<<grid, THREADS, 0, stream>>>(input, labels, weightx, weighty, out);
}